// GNN_15195594293633
// MI455X (gfx1250) — compile-verified
//
#include <hip/hip_runtime.h>
#include <hip/hip_bf16.h>

typedef float v2f __attribute__((ext_vector_type(2)));
typedef float v8f __attribute__((ext_vector_type(8)));

__device__ __forceinline__ void atomAddF(float* p, float v) {
  __hip_atomic_fetch_add(p, v, __ATOMIC_RELAXED, __HIP_MEMORY_SCOPE_AGENT);
}

// ---------------- small utility kernels ----------------

__global__ void fill_f_k(float* __restrict__ p, float v, int n) {
  int i = blockIdx.x * blockDim.x + threadIdx.x;
  if (i < n) p[i] = v;
}

__global__ void edge_deg_k(float* __restrict__ deg, const int* __restrict__ el, int E) {
  int e = blockIdx.x * blockDim.x + threadIdx.x;
  if (e < E) atomAddF(&deg[el[(size_t)E + e]], 1.0f);  // dst half of edge_list
}

__global__ void dinv_k(const float* __restrict__ deg, float* __restrict__ dinv, int n) {
  int i = blockIdx.x * blockDim.x + threadIdx.x;
  if (i < n) {
    float d = deg[i];
    dinv[i] = d > 0.0f ? rsqrtf(d) : 0.0f;
  }
}

// h[i*256+f] = b[f]  (merges zero-init of the aggregation target with +bias)
__global__ void bias_fill_k(float* __restrict__ h, const float* __restrict__ b, int total) {
  int i = blockIdx.x * blockDim.x + threadIdx.x;
  if (i < total) h[i] = b[i & 255];
}

__global__ void relu_k(float* __restrict__ h, int total) {
  int i = blockIdx.x * blockDim.x + threadIdx.x;
  if (i < total) h[i] = fmaxf(h[i], 0.0f);
}

// ---------------- f32 WMMA GEMM: out[n x hid] = H[n x K] @ W[K x hid] ----------------
// grid.x = ceil(n/16); block = 512 threads = 16 waves; wave w owns cols [16w,16w+16)
__global__ __launch_bounds__(512) void gemm_wmma_k(
    const float* __restrict__ H, const float* __restrict__ W,
    float* __restrict__ out, int n, int K, int hid) {
  __shared__ float sA[16 * 256];          // 16 rows x K (K<=256)
  int m0 = blockIdx.x * 16;
  int tot = 16 * K;
  for (int idx = threadIdx.x; idx < tot; idx += blockDim.x) {
    int r = idx / K, c = idx - r * K;
    int row = m0 + r;
    sA[idx] = (row < n) ? H[(size_t)row * K + c] : 0.0f;
  }
  __syncthreads();

  int wave = threadIdx.x >> 5;
  int lane = threadIdx.x & 31;
  int n0   = wave * 16;
  int mr   = lane & 15;                   // M (for A) / N (for B,D)
  int kh   = lane >> 4;                   // K-half select

  v8f c = {0.f, 0.f, 0.f, 0.f, 0.f, 0.f, 0.f, 0.f};
  const float* sArow = sA + mr * K;
  for (int k = 0; k < K; k += 4) {
    int ka = k + 2 * kh;
    v2f a, b;
    // A 16x4 f32 layout: lanes 0-15 -> K={k,k+1}, lanes 16-31 -> K={k+2,k+3}
    a.x = sArow[ka];
    a.y = sArow[ka + 1];
    // B 4x16 f32 layout: VGPR0 rows {k, k+2}, VGPR1 rows {k+1, k+3}; N striped on lanes
    const float* wp = W + (size_t)ka * hid + n0 + mr;
    b.x = wp[0];
    b.y = wp[hid];
    c = __builtin_amdgcn_wmma_f32_16x16x4_f32(false, a, false, b, (short)0, c, false, false);
  }

  // D 16x16 f32: VGPR r -> M = r (lanes 0-15) / 8+r (lanes 16-31); N = lane&15
  int col = n0 + mr;
  int rowbase = m0 + kh * 8;
#pragma unroll
  for (int r = 0; r < 8; ++r) {
    int row = rowbase + r;
    if (row < n) out[(size_t)row * hid + col] = c[r];
  }
}

// ---------------- sparse aggregation: out[dst] += norm * tmp[src] ----------------
// one wave per (edge or self-loop); each lane handles 8 contiguous feats of 256
__global__ void agg_k(const float* __restrict__ tmp, float* __restrict__ out,
                      const int* __restrict__ el, const float* __restrict__ dinv,
                      int E, int n) {
  int gid  = blockIdx.x * blockDim.x + threadIdx.x;
  int e    = gid >> 5;
  int lane = gid & 31;
  if (e >= E + n) return;
  int s, d;
  if (e < E) { s = el[e]; d = el[(size_t)E + e]; }
  else       { s = e - E; d = s; }                 // self loop
  float nrm = dinv[s] * dinv[d];
  const float4* sp = (const float4*)(tmp + (size_t)s * 256) + lane * 2;
  float4 v0 = sp[0];
  float4 v1 = sp[1];
  float* dp = out + (size_t)d * 256 + lane * 8;
  atomAddF(dp + 0, nrm * v0.x); atomAddF(dp + 1, nrm * v0.y);
  atomAddF(dp + 2, nrm * v0.z); atomAddF(dp + 3, nrm * v0.w);
  atomAddF(dp + 4, nrm * v1.x); atomAddF(dp + 5, nrm * v1.y);
  atomAddF(dp + 6, nrm * v1.z); atomAddF(dp + 7, nrm * v1.w);
}

// ---------------- pooling ----------------

__global__ void count_k(float* __restrict__ cnts, const int* __restrict__ batch, int n) {
  int i = blockIdx.x * blockDim.x + threadIdx.x;
  if (i < n) atomAddF(&cnts[batch[i]], 1.0f);
}

// block of 256 threads handles 256 nodes; LDS 16x256 partial sums, then atomics
__global__ __launch_bounds__(256) void pool_k(const float* __restrict__ h,
                                              const int* __restrict__ batch,
                                              float* __restrict__ sums, int n) {
  __shared__ float ls[16 * 256];
  int t = threadIdx.x;
  for (int i = t; i < 16 * 256; i += 256) ls[i] = 0.0f;
  __syncthreads();
  int base = blockIdx.x * 256;
  int lim  = n - base; if (lim > 256) lim = 256;
  for (int j = 0; j < lim; ++j) {
    int i = base + j;
    int g = batch[i];
    ls[g * 256 + t] += h[(size_t)i * 256 + t];     // column t only -> no LDS race
  }
  __syncthreads();
  for (int g = 0; g < 16; ++g) {
    float v = ls[g * 256 + t];
    if (v != 0.0f) atomAddF(&sums[g * 256 + t], v);
  }
}

__global__ void final_k(const float* __restrict__ sums, const float* __restrict__ cnts,
                        const float* __restrict__ Wlin, const float* __restrict__ blin,
                        float* __restrict__ out, int OUT) {
  int idx = blockIdx.x * blockDim.x + threadIdx.x;
  if (idx >= 16 * OUT) return;
  int g = idx / OUT, o = idx - g * OUT;
  float cnt = cnts[g]; cnt = cnt > 1.0f ? cnt : 1.0f;
  float inv = 1.0f / cnt;
  float acc = 0.0f;
  for (int k = 0; k < 256; ++k) acc += sums[g * 256 + k] * inv * Wlin[(size_t)k * OUT + o];
  out[idx] = acc + blin[o];
}

// ---------------- host ----------------

extern "C" void kernel_launch(void* const* d_in, const int* in_sizes, int n_in,
                              void* d_out, int out_size, void* d_ws, size_t ws_size,
                              hipStream_t stream) {
  const float* x    = (const float*)d_in[0];
  const int*   el   = (const int*)  d_in[1];
  const int*   bat  = (const int*)  d_in[2];
  const float* W1   = (const float*)d_in[3];
  const float* b1   = (const float*)d_in[4];
  const float* Wm   = (const float*)d_in[5];
  const float* bm   = (const float*)d_in[6];
  const float* Wlin = (const float*)d_in[7];
  const float* blin = (const float*)d_in[8];

  const int n   = in_sizes[2];             // 20000
  const int E   = in_sizes[1] / 2;         // 320000
  const int K0  = in_sizes[0] / n;         // 128
  const int HID = in_sizes[4];             // 256
  const int NL1 = in_sizes[6] / HID;       // 29 middle layers
  const int OUT = in_sizes[8];             // 32

  char* ws = (char*)d_ws;
  size_t off = 0;
  auto carve = [&](size_t bytes) -> char* {
    char* p = ws + off;
    off += (bytes + 255) & ~(size_t)255;
    return p;
  };
  float* deg  = (float*)carve((size_t)n * 4);
  float* dinv = (float*)carve((size_t)n * 4);
  float* tmp  = (float*)carve((size_t)n * HID * 4);
  float* hA   = (float*)carve((size_t)n * HID * 4);
  float* hB   = (float*)carve((size_t)n * HID * 4);
  float* sums = (float*)carve((size_t)16 * HID * 4);
  float* cnts = (float*)carve(16 * 4);
  (void)ws_size;

  const int T = 256;
  // degree (self-loop counted as 1) and symmetric norm
  fill_f_k<<<(n + T - 1) / T, T, 0, stream>>>(deg, 1.0f, n);
  edge_deg_k<<<(E + T - 1) / T, T, 0, stream>>>(deg, el, E);
  dinv_k<<<(n + T - 1) / T, T, 0, stream>>>(deg, dinv, n);

  const int totalF   = n * HID;
  const int aggTh    = (E + n) * 32;
  auto layer = [&](const float* Hin, int K, const float* W, const float* b, float* Hout) {
    gemm_wmma_k<<<dim3((n + 15) / 16), 512, 0, stream>>>(Hin, W, tmp, n, K, HID);
    bias_fill_k<<<(totalF + T - 1) / T, T, 0, stream>>>(Hout, b, totalF);
    agg_k<<<(aggTh + T - 1) / T, T, 0, stream>>>(tmp, Hout, el, dinv, E, n);
    relu_k<<<(totalF + T - 1) / T, T, 0, stream>>>(Hout, totalF);
  };

  layer(x, K0, W1, b1, hA);
  float* cur = hA;
  float* nxt = hB;
  for (int l = 0; l < NL1; ++l) {
    layer(cur, HID, Wm + (size_t)l * HID * HID, bm + (size_t)l * HID, nxt);
    float* t2 = cur; cur = nxt; nxt = t2;
  }

  // global mean pool + final linear
  fill_f_k<<<1, T, 0, stream>>>(cnts, 0.0f, 16);
  fill_f_k<<<(16 * HID + T - 1) / T, T, 0, stream>>>(sums, 0.0f, 16 * HID);
  count_k<<<(n + T - 1) / T, T, 0, stream>>>(cnts, bat, n);
  pool_k<<<(n + 255) / 256, 256, 0, stream>>>(cur, bat, sums, n);
  final_k<<<(16 * OUT + T - 1) / T, T, 0, stream>>>(sums, cnts, Wlin, blin, (float*)d_out, OUT);
}